// MaxUnpooling2D_45853070852410
// MI455X (gfx1250) — compile-verified
//
#include <hip/hip_runtime.h>

// MaxUnpooling2D scatter-add for MI455X (gfx1250, wave32).
//
// Shapes (static, from reference):
//   vals/idx : [16, 128, 128, 64] f32/i32  -> 16,777,216 elements
//   out      : [16, 256, 256, 64] f32      -> 67,108,864 elements
//   per-batch input elems  = 128*128*64 = 1,048,576 = 2^20
//   per-batch output elems = 256*256*64 = 4,194,304 (HWC)
//
// Memory-bound scatter: no matrix math exists here, so the optimization
// target is the 23.3 TB/s HBM path: b128 vector memory ops, NT temporal
// hints on the streaming reads (protect L2 for the atomic working set),
// and no-return global_atomic_add_f32.

typedef __attribute__((ext_vector_type(4))) float v4f;
typedef __attribute__((ext_vector_type(4))) int   v4i;

#define HWC_OUT   (256 * 256 * 64)   // output elems per batch
#define LOG2_IN_B4 18                // (2^20 input elems per batch) / 4-per-thread = 2^18

__global__ __launch_bounds__(256) void unpool_zero_fill(v4f* __restrict__ out, int n4) {
    int i = blockIdx.x * 256 + threadIdx.x;
    if (i < n4) {
        v4f z = (v4f)0.0f;
        // Plain store is fine; NT keeps the 256MB fill from churning L2 ahead
        // of the scatter pass (output > L2 capacity anyway).
        __builtin_nontemporal_store(z, out + i);
    }
}

__global__ __launch_bounds__(256) void unpool_scatter_add(const v4f* __restrict__ vals,
                                                          const v4i* __restrict__ idx,
                                                          float* __restrict__ out,
                                                          int n4) {
    int i = blockIdx.x * 256 + threadIdx.x;
    if (i >= n4) return;

    // Streaming, read-once: non-temporal b128 loads so they don't evict the
    // per-batch 16MB output region that the atomics below are hammering in L2.
    v4f v  = __builtin_nontemporal_load(vals + i);
    v4i ix = __builtin_nontemporal_load(idx + i);

    // 4 consecutive elements never cross a batch boundary (2^20 % 4 == 0).
    int b = i >> LOG2_IN_B4;
    float* base = out + (size_t)b * (size_t)HWC_OUT;

    // Indices are per-sample flat offsets into H*W*C, guaranteed in range.
    // Result discarded -> lowers to no-return global_atomic_add_f32.
    atomicAdd(base + ix.x, v.x);
    atomicAdd(base + ix.y, v.y);
    atomicAdd(base + ix.z, v.z);
    atomicAdd(base + ix.w, v.w);
}

extern "C" void kernel_launch(void* const* d_in, const int* in_sizes, int n_in,
                              void* d_out, int out_size, void* d_ws, size_t ws_size,
                              hipStream_t stream) {
    const v4f* vals = (const v4f*)d_in[0];
    const v4i* idx  = (const v4i*)d_in[1];
    float*     out  = (float*)d_out;

    const int n_in_elems = in_sizes[0];      // 16,777,216
    const int n4_in      = n_in_elems / 4;   //  4,194,304 threads
    const int n4_out     = out_size / 4;     // 16,777,216 threads

    // 1) Zero the unpooled output (harness poisons it; scatter needs zeros).
    {
        int blocks = (n4_out + 255) / 256;
        unpool_zero_fill<<<blocks, 256, 0, stream>>>((v4f*)d_out, n4_out);
    }
    // 2) Atomic scatter-add of pooled values into the zeroed volume.
    {
        int blocks = (n4_in + 255) / 256;
        unpool_scatter_add<<<blocks, 256, 0, stream>>>(vals, idx, out, n4_in);
    }
}